// PCEN_42949673605
// MI455X (gfx1250) — compile-verified
//
#include <hip/hip_runtime.h>

#define TT      2048
#define NBANDS  128
#define NBATCH  16
#define ROWS    (NBATCH * NBANDS)   // 2048 independent IIR rows
#define CHUNK   16
#define NCHUNKS (TT / CHUNK)        // 128
#define EPSV    1e-5f

typedef __attribute__((ext_vector_type(2))) float v2f;
typedef __attribute__((ext_vector_type(4))) float v4f;
typedef __attribute__((ext_vector_type(8))) float v8f;

__device__ __forceinline__ float s_coef() {
    // s = (sqrt(1 + 4*T_VAL^2) - 1) / (2*T_VAL^2), T_VAL = 256 (f32, as in reference)
    return (sqrtf(262145.0f) - 1.0f) / 131072.0f;
}

__device__ __forceinline__ float qpow(float q, int e) {
    float r = 1.0f;
#pragma unroll 1
    for (int i = 0; i < e; ++i) r *= q;
    return r;
}

// y^p for y guaranteed normal & positive: raw v_log_f32 (log2) + v_exp_f32 (2^x).
__device__ __forceinline__ float fast_pow_pos(float y, float p) {
    return __builtin_amdgcn_exp2f(p * __builtin_amdgcn_logf(y));
}

// ---------------------------------------------------------------------------
// Pass 0: per-band constants (128 bands): a=e^alpha, d=e^delta, rr=e^r, d^rr.
// Removes 4.19M-fold redundant transcendental work from the main pass.
// ---------------------------------------------------------------------------
__global__ void __launch_bounds__(NBANDS)
pcen_pass0_bands(const float* __restrict__ alpha, const float* __restrict__ delta,
                 const float* __restrict__ r, float* __restrict__ bws) {
    int b = threadIdx.x;                  // 0..127
    float a  = __expf(alpha[b]);
    float d  = __expf(delta[b]);
    float rr = __expf(r[b]);
    bws[b]              = a;
    bws[NBANDS + b]     = d;
    bws[2 * NBANDS + b] = rr;
    bws[3 * NBANDS + b] = fast_pow_pos(d, rr);      // d^rr  (d = e^delta > 0, normal)
}

// ---------------------------------------------------------------------------
// Pass 1: chunk-local tail sums  L[row][chunk] = s * sum_{i=0..15} q^(15-i) x[i]
// ---------------------------------------------------------------------------
__global__ void __launch_bounds__(256)
pcen_pass1_chunklocal(const float* __restrict__ x, float* __restrict__ Lws) {
    int gid   = blockIdx.x * blockDim.x + threadIdx.x;   // 0 .. ROWS*NCHUNKS-1
    int row   = gid >> 7;                                // / NCHUNKS
    int chunk = gid & (NCHUNKS - 1);
    const float s = s_coef();
    const float q = 1.0f - s;

    const v4f* p4 = (const v4f*)(x + (size_t)row * TT + (size_t)chunk * CHUNK);
    float acc = 0.0f;
#pragma unroll
    for (int i = 0; i < 4; ++i) {
        v4f v = p4[i];
        acc = acc * q + v.x;
        acc = acc * q + v.y;
        acc = acc * q + v.z;
        acc = acc * q + v.w;
    }
    Lws[gid] = s * acc;   // Horner fold == sum x[i] * q^(15-i)
}

// ---------------------------------------------------------------------------
// Pass 2: per-row carry scan: c_{k+1} = q^16 * c_k + L_k,  c_0 = x[row][0]
// Cws[row][k] = smoother value feeding chunk k (m[t0-1]).
// ---------------------------------------------------------------------------
__global__ void __launch_bounds__(256)
pcen_pass2_carry(const float* __restrict__ x, const float* __restrict__ Lws,
                 float* __restrict__ Cws) {
    int row = blockIdx.x * blockDim.x + threadIdx.x;     // 0..ROWS-1
    const float s  = s_coef();
    const float q  = 1.0f - s;
    const float q2 = q * q, q4 = q2 * q2, q8 = q4 * q4, q16 = q8 * q8;

    float c = x[(size_t)row * TT];                       // m[-1] := x[0]
    const float* Lr = Lws + (size_t)row * NCHUNKS;
    float*       Cr = Cws + (size_t)row * NCHUNKS;
#pragma unroll 4
    for (int k = 0; k < NCHUNKS; ++k) {
        Cr[k] = c;
        c = q16 * c + Lr[k];
    }
}

// ---------------------------------------------------------------------------
// Pass 3: per (16-row group, 16-time chunk) wave:
//   smoother = x_tile(16x16) @ V(16x16, V[i][j]=s*q^(j-i), j>=i) + carry[M]*q^(N+1)
// via 4 chained V_WMMA_F32_16X16X4_F32, then elementwise PCEN + transposed store.
// ---------------------------------------------------------------------------
__global__ void __launch_bounds__(32)
pcen_pass3_wmma(const float* __restrict__ x, const float* __restrict__ Cws,
                const float* __restrict__ bws, float* __restrict__ out) {
    const float s = s_coef();
    const float q = 1.0f - s;

    const int chunk = blockIdx.x;            // 0..127 time chunk
    const int rg    = blockIdx.y;            // 0..127 row group
    const int row0  = rg * 16;
    const int t0    = chunk * CHUNK;

    const int lane = threadIdx.x;            // 0..31 (wave32)
    const int m    = lane & 15;
    const int hi   = lane >> 4;

    // B tile (decay matrix V): VGPRv holds K = kk*4 + 2*hi + v, N = m
    v2f Bm[4];
#pragma unroll
    for (int kk = 0; kk < 4; ++kk) {
        int i0 = kk * 4 + 2 * hi;
        Bm[kk].x = (m >= i0)     ? s * qpow(q, m - i0)     : 0.0f;
        Bm[kk].y = (m >= i0 + 1) ? s * qpow(q, m - i0 - 1) : 0.0f;
    }

    // A tile: lane holds row row0+m, K = kk*4 + 2*hi + {0,1}  (8B-aligned b64 loads)
    const float* xr = x + (size_t)(row0 + m) * TT;
    v2f Am[4];
#pragma unroll
    for (int kk = 0; kk < 4; ++kk) {
        int bt = t0 + kk * 4 + 2 * hi;
        Am[kk] = *(const v2f*)(xr + bt);
    }

    // Accumulator init = carry outer product: C[M][N] = c[M] * q^(N+1)
    // C/D layout: lane holds M = v + 8*hi, N = m
    const float qn1 = qpow(q, m + 1);
    v8f acc;
#pragma unroll
    for (int v = 0; v < 8; ++v) {
        int rowM = row0 + v + 8 * hi;
        acc[v] = Cws[(size_t)rowM * NCHUNKS + chunk] * qn1;
    }

    // 16x16x16 triangular matmul as 4 chained fp32 WMMAs
    acc = __builtin_amdgcn_wmma_f32_16x16x4_f32(false, Am[0], false, Bm[0], (short)0, acc, false, false);
    acc = __builtin_amdgcn_wmma_f32_16x16x4_f32(false, Am[1], false, Bm[1], (short)0, acc, false, false);
    acc = __builtin_amdgcn_wmma_f32_16x16x4_f32(false, Am[2], false, Bm[2], (short)0, acc, false, false);
    acc = __builtin_amdgcn_wmma_f32_16x16x4_f32(false, Am[3], false, Bm[3], (short)0, acc, false, false);

    // Per-band constants: 8 consecutive bands per lane -> 2x b128 loads per table
    const int t         = t0 + m;
    const int bidx      = row0 >> 7;                  // batch index (C==1)
    const int band_base = (row0 & 127) + 8 * hi;      // multiple of 8 -> 32B aligned
    const float* ea  = bws;
    const float* ed  = bws + NBANDS;
    const float* err = bws + 2 * NBANDS;
    const float* edr = bws + 3 * NBANDS;
    v4f a_lo  = *(const v4f*)(ea  + band_base), a_hi  = *(const v4f*)(ea  + band_base + 4);
    v4f d_lo  = *(const v4f*)(ed  + band_base), d_hi  = *(const v4f*)(ed  + band_base + 4);
    v4f r_lo  = *(const v4f*)(err + band_base), r_hi  = *(const v4f*)(err + band_base + 4);
    v4f dr_lo = *(const v4f*)(edr + band_base), dr_hi = *(const v4f*)(edr + band_base + 4);
    float av[8]  = {a_lo.x, a_lo.y, a_lo.z, a_lo.w, a_hi.x, a_hi.y, a_hi.z, a_hi.w};
    float dv[8]  = {d_lo.x, d_lo.y, d_lo.z, d_lo.w, d_hi.x, d_hi.y, d_hi.z, d_hi.w};
    float rv[8]  = {r_lo.x, r_lo.y, r_lo.z, r_lo.w, r_hi.x, r_hi.y, r_hi.z, r_hi.w};
    float drv[8] = {dr_lo.x, dr_lo.y, dr_lo.z, dr_lo.w, dr_hi.x, dr_hi.y, dr_hi.z, dr_hi.w};

    // Elementwise PCEN: lane holds D[M][N], M = v + 8*hi, N = m.
    // log(EPS)+log1p(sm/EPS) == log(EPS+sm); both log args are provably normal
    // (>= 1e-5 and >= e^delta), so raw v_log_f32/v_exp_f32 are exact enough and
    // skip the libm subnormal guards. exp2(p*log2(y)) == y^p, no ln2 rescale.
    float res[8];
#pragma unroll
    for (int v = 0; v < 8; ++v) {
        int   rowM = row0 + v + 8 * hi;
        float xv   = x[(size_t)rowM * TT + t];        // 64B-coalesced per half-wave
        float sm   = fmaxf(acc[v], 0.0f);
        float smooth = fast_pow_pos(EPSV + sm, -av[v]);      // (eps+sm)^(-a)
        float base   = fmaf(xv, smooth, dv[v]);              // >= d = e^delta > 0
        res[v] = fast_pow_pos(base, rv[v]) - drv[v];         // base^rr - d^rr
    }
    float* ob = out + (size_t)bidx * TT * NBANDS + (size_t)t * NBANDS + band_base;
    v4f lo;  lo.x  = res[0]; lo.y  = res[1]; lo.z  = res[2]; lo.w  = res[3];
    v4f hi4; hi4.x = res[4]; hi4.y = res[5]; hi4.z = res[6]; hi4.w = res[7];
    *(v4f*)(ob)     = lo;     // 32B-aligned (band_base % 8 == 0)
    *(v4f*)(ob + 4) = hi4;
}

// ---------------------------------------------------------------------------
extern "C" void kernel_launch(void* const* d_in, const int* in_sizes, int n_in,
                              void* d_out, int out_size, void* d_ws, size_t ws_size,
                              hipStream_t stream) {
    const float* x     = (const float*)d_in[0];
    const float* alpha = (const float*)d_in[1];
    const float* delta = (const float*)d_in[2];
    const float* r     = (const float*)d_in[3];
    float* out = (float*)d_out;

    // workspace: band consts (2 KB) + L (1 MB) + carries (1 MB)
    float* bws = (float*)d_ws;
    float* Lws = bws + 4 * NBANDS;
    float* Cws = Lws + (size_t)ROWS * NCHUNKS;

    pcen_pass0_bands<<<1, NBANDS, 0, stream>>>(alpha, delta, r, bws);
    pcen_pass1_chunklocal<<<(ROWS * NCHUNKS) / 256, 256, 0, stream>>>(x, Lws);
    pcen_pass2_carry<<<ROWS / 256, 256, 0, stream>>>(x, Lws, Cws);

    dim3 g3(NCHUNKS, ROWS / 16);   // 128 x 128 independent wave-tasks
    pcen_pass3_wmma<<<g3, 32, 0, stream>>>(x, Cws, bws, out);
}